// MpcRetrievalLoss_38525856645686
// MI455X (gfx1250) — compile-verified
//
#include <hip/hip_runtime.h>
#include <hip/hip_bf16.h>
#include <cstddef>
#include <cstdint>

// ---------------------------------------------------------------------------
// MpcRetrievalLoss for MI455X (gfx1250, wave32, WMMA f32 16x16x4 + TDM)
// ---------------------------------------------------------------------------

typedef float v2f __attribute__((ext_vector_type(2)));
typedef float v8f __attribute__((ext_vector_type(8)));
typedef unsigned int v4u __attribute__((ext_vector_type(4)));
typedef int v4i __attribute__((ext_vector_type(4)));
typedef int v8i __attribute__((ext_vector_type(8)));

namespace {

constexpr int NROW = 2048;   // N
constexpr int DDIM = 512;    // D
constexpr int KS   = 7;      // K samples
constexpr int DCAT = 1024;   // 2*D concatenated GEMM K-dim
constexpr float LOG_2PI_F = 1.8378770664093453f;
constexpr float VIB_BETA_F = 1e-4f;

__device__ __forceinline__ float wave_sum(float v) {
#pragma unroll
  for (int o = 16; o > 0; o >>= 1) v += __shfl_xor(v, o, 32);
  return v;
}

// ---------------------------------------------------------------------------
// Kernel 1: query-side preprocessing. One wave (32 lanes) per row.
//   Acat[i, 0:512]    = exp(-ls)              (inv_sigmas)
//   Acat[i, 512:1024] = qm * exp(-ls)
//   qconst[i] = sum_d qm^2 * inv
//   normc[i]  = -0.5*D*log(2pi) - 0.5*sum_d ls
//   klq[i]    = per-row partial of query VIB KL
// ---------------------------------------------------------------------------
__global__ __launch_bounds__(256)
void prep_query(const float* __restrict__ qmean, const float* __restrict__ qls,
                const float* __restrict__ qeps,  float* __restrict__ Acat,
                float* __restrict__ qconst, float* __restrict__ normc,
                float* __restrict__ klq) {
  const int wave = threadIdx.x >> 5;
  const int lane = threadIdx.x & 31;
  const int row  = blockIdx.x * 8 + wave;

  const float* mrow = qmean + (size_t)row * DDIM;
  const float* lrow = qls   + (size_t)row * DDIM;

  float m[16], ls[16];
  float sumsq = 0.f;
#pragma unroll
  for (int e = 0; e < 16; ++e) {
    const int d = e * 32 + lane;
    m[e]  = mrow[d];
    ls[e] = lrow[d];
    sumsq += m[e] * m[e];
  }
  sumsq = wave_sum(sumsq);
  const float rn = 1.0f / fmaxf(sqrtf(sumsq), 1e-12f);

  float qc = 0.f, sls = 0.f, kls = 0.f;
#pragma unroll
  for (int e = 0; e < 16; ++e) {
    const int d = e * 32 + lane;
    const float q   = m[e] * rn;           // l2-normalized mean
    const float el  = expf(ls[e]);         // sigma
    const float inv = expf(-ls[e]);        // 1/sigma
    Acat[(size_t)row * DCAT + d]        = inv;
    Acat[(size_t)row * DCAT + DDIM + d] = q * inv;
    qc  += q * q * inv;
    sls += ls[e];
    float eb = 0.f;
    for (int k = 0; k < KS; ++k)
      eb += qeps[((size_t)row * KS + k) * DDIM + d];
    eb *= (1.0f / 7.0f);
    const float mu = q + eb * el;          // mean_k of q_samples
    kls += 1.0f + ls[e] - mu * mu - el;
  }
  qc  = wave_sum(qc);
  sls = wave_sum(sls);
  kls = wave_sum(kls);
  if (lane == 0) {
    qconst[row] = qc;
    normc[row]  = -0.5f * ((float)DDIM * LOG_2PI_F) - 0.5f * sls;
    klq[row]    = -0.5f * kls;
  }
}

// ---------------------------------------------------------------------------
// Kernel 2: target-side preprocessing. One wave per row.
//   Bcat[j, 0:512]    = t2bar  = tm^2 + 2 tm sig ebar + sig^2 e2bar
//   Bcat[j, 512:1024] = -2*tbar, tbar = tm + sig*ebar
//   klt[j] = per-row partial of target VIB KL
// ---------------------------------------------------------------------------
__global__ __launch_bounds__(256)
void prep_target(const float* __restrict__ tmean, const float* __restrict__ tls,
                 const float* __restrict__ teps,  float* __restrict__ Bcat,
                 float* __restrict__ klt) {
  const int wave = threadIdx.x >> 5;
  const int lane = threadIdx.x & 31;
  const int row  = blockIdx.x * 8 + wave;

  const float* mrow = tmean + (size_t)row * DDIM;
  const float* lrow = tls   + (size_t)row * DDIM;

  float m[16], ls[16];
  float sumsq = 0.f;
#pragma unroll
  for (int e = 0; e < 16; ++e) {
    const int d = e * 32 + lane;
    m[e]  = mrow[d];
    ls[e] = lrow[d];
    sumsq += m[e] * m[e];
  }
  sumsq = wave_sum(sumsq);
  const float rn = 1.0f / fmaxf(sqrtf(sumsq), 1e-12f);

  float kls = 0.f;
#pragma unroll
  for (int e = 0; e < 16; ++e) {
    const int d = e * 32 + lane;
    const float t   = m[e] * rn;
    const float sig = expf(ls[e]);
    float eb = 0.f, e2b = 0.f;
    for (int k = 0; k < KS; ++k) {
      const float v = teps[((size_t)row * KS + k) * DDIM + d];
      eb  += v;
      e2b += v * v;
    }
    eb  *= (1.0f / 7.0f);
    e2b *= (1.0f / 7.0f);
    const float tbar  = t + sig * eb;
    const float t2bar = t * t + 2.0f * t * sig * eb + sig * sig * e2b;
    Bcat[(size_t)row * DCAT + d]        = t2bar;
    Bcat[(size_t)row * DCAT + DDIM + d] = -2.0f * tbar;
    kls += 1.0f + ls[e] - tbar * tbar - sig;
  }
  kls = wave_sum(kls);
  if (lane == 0) klt[row] = -0.5f * kls;
}

// ---------------------------------------------------------------------------
// Tensor Data Mover: DMA a 2-D tile (tile_rows x KC fp32, row stride DCAT)
// from global into LDS, inserting 4 DWORDs of padding after every 32 DWORDs
// so LDS rows land at stride 36 floats (bank-conflict-free fragment reads).
// D# built per CDNA5 ISA 8.3/8.4. Tracked with TENSORcnt.
// ---------------------------------------------------------------------------
constexpr int BM  = 128;   // block rows (i)
constexpr int BN  = 16;    // block cols (j)
constexpr int KC  = 32;    // K-chunk (floats)
constexpr int LDA = 36;    // padded LDS row stride = KC + 4
constexpr int NCHUNK = DCAT / KC;

__device__ __forceinline__ void tdm_load_tile(uint32_t lds_byte_off,
                                              const float* gptr,
                                              int tile_rows) {
  const uint64_t ga = (uint64_t)(uintptr_t)gptr;
  v4u g0;
  g0[0] = 1u;                                   // count=1, user descriptor
  g0[1] = lds_byte_off;                         // lds_addr (bytes)
  g0[2] = (uint32_t)ga;                         // global_addr[31:0]
  g0[3] = (uint32_t)((ga >> 32) & 0x01FFFFFFu)  // global_addr[56:32]
        | (2u << 30);                           // type = 2 ("image")
  v8i g1;
  g1[0] = (int)((2u << 16)      // data_size = 4 bytes
              | (1u << 20)      // pad_enable
              | (4u << 22)      // pad_interval: 1<<4 * 8B = 32 DWORDs
              | (3u << 25));    // pad_amount: 3+1 = 4 DWORDs
  g1[1] = (int)(((uint32_t)DCAT & 0xFFFFu) << 16);  // tensor_dim0 lo16
  g1[2] = (int)(((uint32_t)NROW & 0xFFFFu) << 16);  // dim0 hi16=0 | tensor_dim1 lo16
  g1[3] = (int)((uint32_t)KC << 16);                // dim1 hi16=0 | tile_dim0 = KC
  g1[4] = tile_rows;                                // tile_dim1 | tile_dim2=0
  g1[5] = DCAT;                                     // tensor_dim0_stride lo32
  g1[6] = 0;                                        // stride hi16 | dim1_stride lo16
  g1[7] = 0;
  const v4i z4 = {0, 0, 0, 0};
#if defined(__clang_major__) && (__clang_major__ >= 23)
  const v8i z8 = {0, 0, 0, 0, 0, 0, 0, 0};
  __builtin_amdgcn_tensor_load_to_lds(g0, g1, z4, z4, z8, 0);
#else
  __builtin_amdgcn_tensor_load_to_lds(g0, g1, z4, z4, 0);
#endif
}

// ---------------------------------------------------------------------------
// Kernel 3: GEMM -> scores.  scores[i,j] = -0.5*(Acat[i].Bcat[j] + qconst[i])
//                                          + normc[j]
// Block = 256 threads (8 waves). Block tile: 128 rows (i) x 16 cols (j).
// Each wave owns one 16x16 WMMA f32 tile; Bs fragment shared by all 8 waves.
// Wave 0 drives the TDM, double-buffered: chunk k+1 streams into the other
// LDS buffer while all waves run WMMA on chunk k.  s_wait_tensorcnt 2 lets
// the two just-issued DMAs stay in flight (in-order completion guarantees
// the chunk being consumed has landed).
// ---------------------------------------------------------------------------
__global__ __launch_bounds__(256)
void gemm_scores(const float* __restrict__ Acat, const float* __restrict__ Bcat,
                 const float* __restrict__ qconst, const float* __restrict__ normc,
                 float* __restrict__ scores) {
  __shared__ float As[2][BM * LDA];   // 2*128*36*4 = 36864 B
  __shared__ float Bs[2][BN * LDA];   // 2* 16*36*4 =  4608 B

  const int tid  = threadIdx.x;
  const int wave = tid >> 5;
  const int lane = tid & 31;
  const int lrow = lane & 15;      // fragment row (A) / col (B)
  const int half = lane >> 4;      // K half: lanes 0-15 -> k{0,1}, 16-31 -> k{2,3}

  const int i0 = blockIdx.x * BM;
  const int j0 = blockIdx.y * BN;

  // LDS aperture keeps the workgroup-relative byte offset in addr[31:0].
  const uint32_t as_off = (uint32_t)(uintptr_t)(const void*)&As[0][0];
  const uint32_t bs_off = (uint32_t)(uintptr_t)(const void*)&Bs[0][0];
  const uint32_t as_buf_bytes = (uint32_t)(BM * LDA * sizeof(float));
  const uint32_t bs_buf_bytes = (uint32_t)(BN * LDA * sizeof(float));

  v8f c = {};

  if (wave == 0) {   // prologue: stream chunk 0 into buffer 0
    tdm_load_tile(as_off, &Acat[(size_t)i0 * DCAT], BM);
    tdm_load_tile(bs_off, &Bcat[(size_t)j0 * DCAT], BN);
  }

  for (int it = 0; it < NCHUNK; ++it) {
    const int buf = it & 1;
    if (wave == 0) {
      if (it + 1 < NCHUNK) {
        const int nb = (it + 1) & 1;
        tdm_load_tile(as_off + (uint32_t)nb * as_buf_bytes,
                      &Acat[(size_t)i0 * DCAT + (it + 1) * KC], BM);
        tdm_load_tile(bs_off + (uint32_t)nb * bs_buf_bytes,
                      &Bcat[(size_t)j0 * DCAT + (it + 1) * KC], BN);
        __builtin_amdgcn_s_wait_tensorcnt((short)2);
      } else {
        __builtin_amdgcn_s_wait_tensorcnt((short)0);
      }
    }
    __syncthreads();   // chunk `it` now resident in As/Bs[buf]

    const float* apW = &As[buf][(wave * 16 + lrow) * LDA + 2 * half];
    const float* bpW = &Bs[buf][lrow * LDA + 2 * half];
#pragma unroll
    for (int kk = 0; kk < KC; kk += 4) {
      v2f a = *(const v2f*)(apW + kk);   // A[m=lrow][k+2*half .. +1]
      v2f b = *(const v2f*)(bpW + kk);   // B[k+2*half][n=lrow]
      c = __builtin_amdgcn_wmma_f32_16x16x4_f32(
          /*neg_a=*/false, a, /*neg_b=*/false, b,
          /*c_mod=*/(short)0, c, /*reuse_a=*/false, /*reuse_b=*/false);
    }
    __syncthreads();   // everyone done with buf before TDM refills it
  }

  // Epilogue: C/D layout — lanes 0-15: M=r, N=lane; lanes 16-31: M=r+8, N=lane-16
  const int jn = j0 + lrow;
  const float nc = normc[jn];
#pragma unroll
  for (int r = 0; r < 8; ++r) {
    const int im = i0 + wave * 16 + r + 8 * half;
    scores[(size_t)im * NROW + jn] = -0.5f * (c[r] + qconst[im]) + nc;
  }
}

// ---------------------------------------------------------------------------
// Kernel 4: column max  m[j] = max_i scores[i,j]  (coalesced across j)
// ---------------------------------------------------------------------------
__global__ __launch_bounds__(256)
void colmax_kernel(const float* __restrict__ scores, float* __restrict__ cmax) {
  const int j = blockIdx.x * 256 + threadIdx.x;
  float m = -3.402823466e38f;
  for (int i = 0; i < NROW; ++i)
    m = fmaxf(m, scores[(size_t)i * NROW + j]);
  cmax[j] = m;
}

// ---------------------------------------------------------------------------
// Kernel 5: per-row stable log-sum-exp of (scores - cmax), diag CE term.
// ---------------------------------------------------------------------------
__global__ __launch_bounds__(256)
void row_lse_kernel(const float* __restrict__ scores, const float* __restrict__ cmax,
                    float* __restrict__ ce) {
  const int i   = blockIdx.x;
  const int tid = threadIdx.x;
  __shared__ float red[256];
  const float* srow = scores + (size_t)i * NROW;

  float m = -3.402823466e38f;
  for (int j = tid; j < NROW; j += 256)
    m = fmaxf(m, srow[j] - cmax[j]);
  red[tid] = m;
  __syncthreads();
  for (int s = 128; s > 0; s >>= 1) {
    if (tid < s) red[tid] = fmaxf(red[tid], red[tid + s]);
    __syncthreads();
  }
  const float rm = red[0];
  __syncthreads();

  float acc = 0.f;
  for (int j = tid; j < NROW; j += 256)
    acc += expf(srow[j] - cmax[j] - rm);
  red[tid] = acc;
  __syncthreads();
  for (int s = 128; s > 0; s >>= 1) {
    if (tid < s) red[tid] += red[tid + s];
    __syncthreads();
  }
  if (tid == 0) {
    const float lse = rm + logf(red[0]);
    ce[i] = lse - (srow[i] - cmax[i]);   // -logp of diagonal
  }
}

// ---------------------------------------------------------------------------
// Kernel 6: deterministic final reduction -> scalar loss
// ---------------------------------------------------------------------------
__global__ __launch_bounds__(256)
void final_reduce(const float* __restrict__ ce, const float* __restrict__ klq,
                  const float* __restrict__ klt, float* __restrict__ out) {
  __shared__ float r0[256];
  __shared__ float r1[256];
  const int tid = threadIdx.x;
  float a = 0.f, b = 0.f;
  for (int i = tid; i < NROW; i += 256) {
    a += ce[i];
    b += klq[i] + klt[i];
  }
  r0[tid] = a;
  r1[tid] = b;
  __syncthreads();
  for (int s = 128; s > 0; s >>= 1) {
    if (tid < s) { r0[tid] += r0[tid + s]; r1[tid] += r1[tid + s]; }
    __syncthreads();
  }
  if (tid == 0)
    out[0] = r0[0] / (float)NROW + VIB_BETA_F * r1[0];
}

}  // namespace

extern "C" void kernel_launch(void* const* d_in, const int* in_sizes, int n_in,
                              void* d_out, int out_size, void* d_ws, size_t ws_size,
                              hipStream_t stream) {
  const float* q_mean = (const float*)d_in[0];
  const float* q_ls   = (const float*)d_in[1];
  const float* t_mean = (const float*)d_in[2];
  const float* t_ls   = (const float*)d_in[3];
  const float* q_eps  = (const float*)d_in[4];
  const float* t_eps  = (const float*)d_in[5];
  float* out = (float*)d_out;

  float* ws     = (float*)d_ws;
  float* Acat   = ws;                                  // 2048*1024
  float* Bcat   = Acat + (size_t)NROW * DCAT;          // 2048*1024
  float* scores = Bcat + (size_t)NROW * DCAT;          // 2048*2048
  float* qconst = scores + (size_t)NROW * NROW;        // 2048
  float* normc  = qconst + NROW;                       // 2048
  float* cmax   = normc + NROW;                        // 2048
  float* klq    = cmax + NROW;                         // 2048
  float* klt    = klq + NROW;                          // 2048
  float* ce     = klt + NROW;                          // 2048

  prep_query<<<NROW / 8, 256, 0, stream>>>(q_mean, q_ls, q_eps, Acat, qconst, normc, klq);
  prep_target<<<NROW / 8, 256, 0, stream>>>(t_mean, t_ls, t_eps, Bcat, klt);

  dim3 ggrid(NROW / BM, NROW / BN);   // (16, 128)
  gemm_scores<<<ggrid, 256, 0, stream>>>(Acat, Bcat, qconst, normc, scores);

  colmax_kernel<<<NROW / 256, 256, 0, stream>>>(scores, cmax);
  row_lse_kernel<<<NROW, 256, 0, stream>>>(scores, cmax, ce);
  final_reduce<<<1, 256, 0, stream>>>(ce, klq, klt, out);
}